// DynamicNeuralTuringMachineMemory_81243601371257
// MI455X (gfx1250) — compile-verified
//
#include <hip/hip_runtime.h>
#include <math.h>

// ---------------- problem constants ----------------
#define N_LOC   8192
#define C_SIZE  2048
#define A_SIZE  512
#define CTRL_IN 1024
#define D_FULL  2560            // A_SIZE + C_SIZE
#define EPSV    1e-7f

// ---------------- d_out layout (floats, concat in return order) ----------------
#define OUT_READVEC 0
#define OUT_NEWMEM  (D_FULL)                       // 2560
#define OUT_ADDR    (D_FULL + N_LOC * C_SIZE)      // 2560 + 16777216
#define OUT_EMA     (OUT_ADDR + N_LOC)

// ---------------- workspace layout (floats) ----------------
#define WS_SC    0              // [0]=beta [1]=gamma [2]=alpha [3]=q_norm
#define WS_Q     16             // 2560 : query vector
#define WS_Z     (WS_Q + D_FULL)        // 8192 : softmax logits
#define WS_DELTA (WS_Z + N_LOC)         // 2048 : cand - erase
#define WS_PART  (WS_DELTA + C_SIZE)    // 64*2560 : read_vec partials
#define NPART    64

typedef __attribute__((ext_vector_type(2))) float v2f;
typedef __attribute__((ext_vector_type(4))) float v4f;
typedef __attribute__((ext_vector_type(8))) float v8f;

// ---------------- deterministic block reductions (wave32) ----------------
__device__ __forceinline__ float waveSum(float v) {
#pragma unroll
  for (int off = 16; off > 0; off >>= 1) v += __shfl_down(v, off, 32);
  return v;
}
__device__ __forceinline__ float waveMax(float v) {
#pragma unroll
  for (int off = 16; off > 0; off >>= 1) v = fmaxf(v, __shfl_down(v, off, 32));
  return v;
}
// result valid on thread 0; smem needs >= 32 floats
__device__ __forceinline__ float blockSum(float v, float* sm) {
  const int lane = threadIdx.x & 31, wid = threadIdx.x >> 5;
  const int nw = blockDim.x >> 5;
  v = waveSum(v);
  if (lane == 0) sm[wid] = v;
  __syncthreads();
  v = (threadIdx.x < nw) ? sm[threadIdx.x] : 0.0f;
  if (wid == 0) v = waveSum(v);
  __syncthreads();
  return v;
}
__device__ __forceinline__ float blockMax(float v, float* sm) {
  const int lane = threadIdx.x & 31, wid = threadIdx.x >> 5;
  const int nw = blockDim.x >> 5;
  v = waveMax(v);
  if (lane == 0) sm[wid] = v;
  __syncthreads();
  v = (threadIdx.x < nw) ? sm[threadIdx.x] : -3.4e38f;
  if (wid == 0) v = waveMax(v);
  __syncthreads();
  return v;
}

__device__ __forceinline__ float dot4(v4f a, v4f b) {
  return a.x * b.x + a.y * b.y + a.z * b.z + a.w * b.w;
}

// ---------------- 1) scalar gates: beta, gamma, alpha ----------------
__global__ void k_scalars(const float* __restrict__ h,  const float* __restrict__ x,
                          const float* __restrict__ us, const float* __restrict__ bs,
                          const float* __restrict__ ul, const float* __restrict__ bl,
                          const float* __restrict__ uca,const float* __restrict__ bca,
                          float* __restrict__ ws) {
  __shared__ float sm[32];
  float d0 = 0.f, d1 = 0.f, d2 = 0.f;
  for (int i = threadIdx.x; i < N_LOC; i += blockDim.x) {
    float hv = h[i];
    d0 += us[i] * hv; d1 += ul[i] * hv; d2 += uca[i] * hv;
  }
  for (int i = threadIdx.x; i < CTRL_IN; i += blockDim.x)
    d2 += uca[N_LOC + i] * x[i];
  d0 = blockSum(d0, sm);
  d1 = blockSum(d1, sm);
  d2 = blockSum(d2, sm);
  if (threadIdx.x == 0) {
    float sb   = d0 + bs[0];
    float beta = (sb > 20.f ? sb : log1pf(expf(sb))) + 1.0f;   // softplus + 1
    float gam  = 1.f / (1.f + expf(-(d1 + bl[0])));            // sigmoid
    ws[WS_SC + 0] = beta;
    ws[WS_SC + 1] = gam;
    ws[WS_SC + 2] = d2 + bca[0];                               // alpha
  }
}

// ---------------- 2) query = W_query @ h + b_query (NT weight stream, b128) -------
__global__ void k_query(const float* __restrict__ Wq, const float* __restrict__ h,
                        const float* __restrict__ bq, float* __restrict__ ws) {
  __shared__ float sm[32];
  const int r = blockIdx.x;
  const v4f* wrow = reinterpret_cast<const v4f*>(Wq + (size_t)r * N_LOC);
  const v4f* h4   = reinterpret_cast<const v4f*>(h);
  float acc = 0.f;
  for (int i = threadIdx.x; i < N_LOC / 4; i += 256) {
    v4f w = __builtin_nontemporal_load(wrow + i);   // single-use weights: bypass L2
    acc += dot4(w, h4[i]);
  }
  acc = blockSum(acc, sm);
  if (threadIdx.x == 0) ws[WS_Q + r] = acc + bq[r];
}

// ---------------- 3) q_norm ----------------
__global__ void k_qnorm(float* __restrict__ ws) {
  __shared__ float sm[32];
  const float* q = ws + WS_Q;
  float acc = 0.f;
  for (int i = threadIdx.x; i < D_FULL; i += blockDim.x) { float v = q[i]; acc += v * v; }
  acc = blockSum(acc, sm);
  if (threadIdx.x == 0) ws[WS_SC + 3] = fmaxf(sqrtf(acc), EPSV);
}

// ---------------- 4) per-row dots + norms -> sim, logits, new_ema ----------------
// full_mem rows are read with RT policy so they stay L2-resident for k_rvpart/k_newmem.
__global__ void k_rows(const float* __restrict__ maddr, const float* __restrict__ mcont,
                       const float* __restrict__ ema, float* __restrict__ ws,
                       float* __restrict__ out) {
  __shared__ float sm[32];
  const int i = blockIdx.x;
  const v4f* q4    = reinterpret_cast<const v4f*>(ws + WS_Q);
  const v4f* arow4 = reinterpret_cast<const v4f*>(maddr + (size_t)i * A_SIZE);
  const v4f* crow4 = reinterpret_cast<const v4f*>(mcont + (size_t)i * C_SIZE);
  float dot = 0.f, nn = 0.f;
  for (int j = threadIdx.x; j < D_FULL / 4; j += 256) {
    v4f v = (j < A_SIZE / 4) ? arow4[j] : crow4[j - A_SIZE / 4];
    dot += dot4(v, q4[j]);
    nn  += dot4(v, v);
  }
  dot = blockSum(dot, sm);
  nn  = blockSum(nn, sm);
  if (threadIdx.x == 0) {
    float beta = ws[WS_SC + 0], gam = ws[WS_SC + 1], qn = ws[WS_SC + 3];
    float mn  = fmaxf(sqrtf(nn), EPSV);
    float sim = beta * dot / (mn * qn);
    float e   = ema[i];
    ws[WS_Z + i]     = sim - gam * e;          // softmax logit
    out[OUT_EMA + i] = 0.1f * e + 0.9f * sim;  // new_ema output
  }
}

// ---------------- 5) softmax over 8192 logits -> address ----------------
__global__ void k_softmax(float* __restrict__ ws, float* __restrict__ out) {
  __shared__ float sm[34];
  const float* z = ws + WS_Z;
  float m = -3.4e38f;
  for (int i = threadIdx.x; i < N_LOC; i += blockDim.x) m = fmaxf(m, z[i]);
  m = blockMax(m, sm);
  if (threadIdx.x == 0) sm[32] = m;
  __syncthreads();
  m = sm[32];
  float s = 0.f;
  for (int i = threadIdx.x; i < N_LOC; i += blockDim.x) s += expf(z[i] - m);
  s = blockSum(s, sm);
  if (threadIdx.x == 0) sm[33] = s;
  __syncthreads();
  s = sm[33];
  for (int i = threadIdx.x; i < N_LOC; i += blockDim.x)
    out[OUT_ADDR + i] = expf(z[i] - m) / s;
}

// ---------------- 6) read_vec partials over 64 row-partitions (rows 0..8190) ------
// full_mem re-read: expected to hit L2 (resident from k_rows).
__global__ void k_rvpart(const float* __restrict__ maddr, const float* __restrict__ mcont,
                         const float* __restrict__ out, float* __restrict__ ws) {
  const int j  = blockIdx.x * 256 + threadIdx.x;           // 0..2559
  const int i0 = blockIdx.y * (N_LOC / NPART);
  int i1 = i0 + (N_LOC / NPART);
  if (i1 > N_LOC - 1) i1 = N_LOC - 1;                      // exclude last row
  const float* addr = out + OUT_ADDR;
  float acc = 0.f;
  for (int i = i0; i < i1; ++i) {
    float a = addr[i];
    float v = (j < A_SIZE) ? maddr[(size_t)i * A_SIZE + j]
                           : mcont[(size_t)i * C_SIZE + (j - A_SIZE)];
    acc += a * v;
  }
  ws[WS_PART + blockIdx.y * D_FULL + j] = acc;
}

__global__ void k_rvreduce(const float* __restrict__ ws, float* __restrict__ out) {
  const int j = blockIdx.x * 256 + threadIdx.x;
  float s = 0.f;
  for (int p = 0; p < NPART; ++p) s += ws[WS_PART + p * D_FULL + j];
  out[OUT_READVEC + j] = s;
}

// ---------------- 7) delta = relu(Wch@h + alpha*(Wci@x)) - (We@h + be) ------------
// All three weight matrices are single-use: non-temporal b128 streams.
__global__ void k_delta(const float* __restrict__ We,  const float* __restrict__ be,
                        const float* __restrict__ Wch, const float* __restrict__ Wci,
                        const float* __restrict__ h,   const float* __restrict__ x,
                        float* __restrict__ ws) {
  __shared__ float sm[32];
  const int j = blockIdx.x;
  const v4f* er = reinterpret_cast<const v4f*>(We  + (size_t)j * N_LOC);
  const v4f* cr = reinterpret_cast<const v4f*>(Wch + (size_t)j * N_LOC);
  const v4f* ir = reinterpret_cast<const v4f*>(Wci + (size_t)j * CTRL_IN);
  const v4f* h4 = reinterpret_cast<const v4f*>(h);
  const v4f* x4 = reinterpret_cast<const v4f*>(x);
  float e = 0.f, c1 = 0.f, c2 = 0.f;
  for (int i = threadIdx.x; i < N_LOC / 4; i += 256) {
    v4f hv = h4[i];
    e  += dot4(__builtin_nontemporal_load(er + i), hv);
    c1 += dot4(__builtin_nontemporal_load(cr + i), hv);
  }
  for (int i = threadIdx.x; i < CTRL_IN / 4; i += 256)
    c2 += dot4(__builtin_nontemporal_load(ir + i), x4[i]);
  e  = blockSum(e,  sm);
  c1 = blockSum(c1, sm);
  c2 = blockSum(c2, sm);
  if (threadIdx.x == 0) {
    float alpha = ws[WS_SC + 2];
    float erase = e + be[j];
    float cand  = fmaxf(c1 + alpha * c2, 0.f);   // relu
    ws[WS_DELTA + j] = cand - erase;
  }
}

// ---------------- 8) new_mem = memory_contents + addr (x) delta via f32 WMMA ------
// Rank-1 as D = A x B + C with K=4: A has addr in column K=0 (VGPR0, lanes 0-15 per
// documented 16x4 f32 A layout), B has delta in row K=0 (VGPR0, lanes 0-15), all
// other K slots zero => exact fp32 addr[m]*delta[n] + mem[m][n]. EXEC all-ones.
// C loads hit L2 (resident from k_rows); D stores are non-temporal (write-once).
__global__ void k_newmem(const float* __restrict__ mcont, const float* __restrict__ addr,
                         const float* __restrict__ delta, float* __restrict__ out) {
  const int w    = (blockIdx.x * blockDim.x + threadIdx.x) >> 5;  // global wave id
  const int lane = threadIdx.x & 31;
  const int tile_n = (w & ((C_SIZE / 16) - 1)) << 4;   // 128 tiles along columns
  const int tile_m = (w >> 7) << 4;                    // 512 tiles along rows

  v2f a, b;
  a.x = (lane < 16) ? addr[tile_m + lane]  : 0.0f;  a.y = 0.0f;
  b.x = (lane < 16) ? delta[tile_n + lane] : 0.0f;  b.y = 0.0f;

  const int half = lane >> 4, nl = lane & 15;
  const int col  = tile_n + nl;
  v8f c;
#pragma unroll
  for (int r = 0; r < 8; ++r) {
    int row = tile_m + half * 8 + r;
    c[r] = mcont[(size_t)row * C_SIZE + col];
  }
  v8f d = __builtin_amdgcn_wmma_f32_16x16x4_f32(
      /*neg_a=*/false, a, /*neg_b=*/false, b,
      /*c_mod=*/(short)0, c, /*reuse_a=*/false, /*reuse_b=*/false);
#pragma unroll
  for (int r = 0; r < 8; ++r) {
    int row = tile_m + half * 8 + r;
    __builtin_nontemporal_store(d[r], &out[(size_t)row * C_SIZE + col]);
  }
}

// ---------------- launch ----------------
extern "C" void kernel_launch(void* const* d_in, const int* in_sizes, int n_in,
                              void* d_out, int out_size, void* d_ws, size_t ws_size,
                              hipStream_t stream) {
  const float* h     = (const float*)d_in[0];
  const float* x     = (const float*)d_in[1];
  const float* mcont = (const float*)d_in[2];
  const float* maddr = (const float*)d_in[3];
  const float* ema   = (const float*)d_in[4];
  const float* Wq    = (const float*)d_in[5];
  const float* bq    = (const float*)d_in[6];
  const float* us    = (const float*)d_in[7];
  const float* bs    = (const float*)d_in[8];
  const float* ul    = (const float*)d_in[9];
  const float* bl    = (const float*)d_in[10];
  const float* We    = (const float*)d_in[11];
  const float* be    = (const float*)d_in[12];
  const float* Wch   = (const float*)d_in[13];
  const float* Wci   = (const float*)d_in[14];
  const float* uca   = (const float*)d_in[15];
  const float* bca   = (const float*)d_in[16];
  float* out = (float*)d_out;
  float* ws  = (float*)d_ws;

  k_scalars <<<1, 1024, 0, stream>>>(h, x, us, bs, ul, bl, uca, bca, ws);
  k_query   <<<D_FULL, 256, 0, stream>>>(Wq, h, bq, ws);
  k_qnorm   <<<1, 1024, 0, stream>>>(ws);
  k_rows    <<<N_LOC, 256, 0, stream>>>(maddr, mcont, ema, ws, out);
  k_softmax <<<1, 1024, 0, stream>>>(ws, out);
  k_rvpart  <<<dim3(D_FULL / 256, NPART), 256, 0, stream>>>(maddr, mcont, out, ws);
  k_rvreduce<<<D_FULL / 256, 256, 0, stream>>>(ws, out);
  k_delta   <<<C_SIZE, 256, 0, stream>>>(We, be, Wch, Wci, h, x, ws);
  k_newmem  <<<(N_LOC / 16) * (C_SIZE / 16) / 8, 256, 0, stream>>>(
      mcont, out + OUT_ADDR, ws + WS_DELTA, out + OUT_NEWMEM);
}